// QuantumEncoder_65481071401688
// MI455X (gfx1250) — compile-verified
//
#include <hip/hip_runtime.h>
#include <math.h>

typedef __attribute__((ext_vector_type(2))) float v2f;
typedef __attribute__((ext_vector_type(8))) float v8f;

#define N_WIRES 4
#define N_LAYERS 3

struct Cpx { float re, im; };
__device__ __forceinline__ Cpx cmul(Cpx a, Cpx b) {
    return { a.re * b.re - a.im * b.im, a.re * b.im + a.im * b.re };
}

// ---------------------------------------------------------------------------
// Setup kernel: build the real 32x32 matrix W from params.
// V = product of variational gates (16x16 complex unitary). Thread j (j<16)
// evolves basis column e_j through the circuit; then
//   y = V s  with interleaved real layout  f = 2j + {0:Re, 1:Im}:
//   W[2j  ][2k  ] =  Re V[k][j]   W[2j  ][2k+1] =  Im V[k][j]
//   W[2j+1][2k  ] = -Im V[k][j]   W[2j+1][2k+1] =  Re V[k][j]
// ---------------------------------------------------------------------------
__global__ void build_W_kernel(const float* __restrict__ params,
                               float* __restrict__ W) {
    const int j = threadIdx.x;
    if (j >= 16) return;

    float cr[16], ci[16];
#pragma unroll
    for (int k = 0; k < 16; ++k) { cr[k] = (k == j) ? 1.0f : 0.0f; ci[k] = 0.0f; }

#pragma unroll
    for (int l = 0; l < N_LAYERS; ++l) {
#pragma unroll
        for (int q = 0; q < N_WIRES; ++q) {
            const float phi = params[(l * 4 + q) * 3 + 0];
            const float th  = params[(l * 4 + q) * 3 + 1];
            const float om  = params[(l * 4 + q) * 3 + 2];
            const int bit = 1 << (3 - q);  // wire 0 = MSB

            // RZ(phi): phase e^{-i phi/2} on bit==0, e^{+i phi/2} on bit==1
            {
                const float c = cosf(0.5f * phi), s0 = sinf(0.5f * phi);
#pragma unroll
                for (int k = 0; k < 16; ++k) {
                    const float s = (k & bit) ? s0 : -s0;
                    const float r  = cr[k] * c - ci[k] * s;
                    const float im = cr[k] * s + ci[k] * c;
                    cr[k] = r; ci[k] = im;
                }
            }
            // RY(th)
            {
                const float ct = cosf(0.5f * th), st = sinf(0.5f * th);
#pragma unroll
                for (int k = 0; k < 16; ++k) {
                    if (k & bit) continue;
                    const int k1 = k | bit;
                    const float r0 = ct * cr[k] - st * cr[k1];
                    const float i0 = ct * ci[k] - st * ci[k1];
                    const float r1 = st * cr[k] + ct * cr[k1];
                    const float i1 = st * ci[k] + ct * ci[k1];
                    cr[k]  = r0; ci[k]  = i0;
                    cr[k1] = r1; ci[k1] = i1;
                }
            }
            // RZ(om)
            {
                const float c = cosf(0.5f * om), s0 = sinf(0.5f * om);
#pragma unroll
                for (int k = 0; k < 16; ++k) {
                    const float s = (k & bit) ? s0 : -s0;
                    const float r  = cr[k] * c - ci[k] * s;
                    const float im = cr[k] * s + ci[k] * c;
                    cr[k] = r; ci[k] = im;
                }
            }
        }
        // CNOT chain: control q, target q+1
#pragma unroll
        for (int q = 0; q < N_WIRES - 1; ++q) {
            const int bc = 1 << (3 - q), bt = 1 << (2 - q);
#pragma unroll
            for (int k = 0; k < 16; ++k) {
                if ((k & bc) && !(k & bt)) {
                    const int k1 = k | bt;
                    const float tr = cr[k], ti = ci[k];
                    cr[k] = cr[k1]; ci[k] = ci[k1];
                    cr[k1] = tr;    ci[k1] = ti;
                }
            }
        }
    }

#pragma unroll
    for (int k = 0; k < 16; ++k) {
        W[(2 * j) * 32 + 2 * k]         =  cr[k];
        W[(2 * j) * 32 + 2 * k + 1]     =  ci[k];
        W[(2 * j + 1) * 32 + 2 * k]     = -ci[k];
        W[(2 * j + 1) * 32 + 2 * k + 1] =  cr[k];
    }
}

// ---------------------------------------------------------------------------
// Main kernel: one wave handles 16 batch rows per tile iteration.
//   Yt(32x16) = Wt(32x32) x St(32x16)  via 2x8 V_WMMA_F32_16X16X4_F32
//   A fragment (Wt tile m, chunk kc): lane(r,h): M=16m+r, K = 4kc+2h+{0,1}
//   B fragment (St chunk kc):          lane(r,h): N=r,    K = 4kc+2h+{0,1}
//     -> feature f = 2(2kc+h)+{0,1} = Re/Im of s_j, j = 2kc+h
//     -> s_j = u0[kc>>2] * u1[(kc>>1)&1] * u2[kc&1] * u3[h]
// D layout: lane(r,h) holds Yt rows {8h..8h+7} (tile0) and {16+8h..} (tile1)
// of batch column r  => all 32 features of one batch row live in lane pair
// (r, r+16); one shfl_xor(16) finishes the Z-expectation reductions.
// ---------------------------------------------------------------------------
__global__ __launch_bounds__(256) void qsim_kernel(const float* __restrict__ x,
                                                   const float* __restrict__ W,
                                                   float* __restrict__ out,
                                                   int numTiles) {
    const int lane = threadIdx.x & 31;
    const int wave = threadIdx.x >> 5;
    const int r = lane & 15;
    const int h = lane >> 4;

    // Preload W^T A-fragments (shared across all tiles): 2 tiles x 8 K-chunks
    v2f aw[2][8];
#pragma unroll
    for (int m = 0; m < 2; ++m) {
#pragma unroll
        for (int kc = 0; kc < 8; ++kc) {
            const int fin  = kc * 4 + 2 * h;
            const int fout = m * 16 + r;
            aw[m][kc].x = W[fin * 32 + fout];
            aw[m][kc].y = W[(fin + 1) * 32 + fout];
        }
    }

    const int wavesPerGrid = gridDim.x * (blockDim.x >> 5);
    for (int tile = blockIdx.x * (blockDim.x >> 5) + wave; tile < numTiles;
         tile += wavesPerGrid) {
        const int batch = tile * 16 + r;

        // ---- encoding: product state u0..u3 for batch row r ----
        const float4 xv = ((const float4*)x)[batch];
        float cs[4], sn[4];
        cs[0] = __cosf(0.5f * xv.x); sn[0] = __sinf(0.5f * xv.x);
        cs[1] = __cosf(0.5f * xv.y); sn[1] = __sinf(0.5f * xv.y);
        cs[2] = __cosf(0.5f * xv.z); sn[2] = __sinf(0.5f * xv.z);
        cs[3] = __cosf(0.5f * xv.w); sn[3] = __sinf(0.5f * xv.w);

        Cpx u[4][2];
#pragma unroll
        for (int i = 0; i < 4; ++i) {
            const float c1 = cs[i],           s1 = sn[i];
            const float c2 = cs[(i + 1) & 3], s2 = sn[(i + 1) & 3];
            const float c3 = cs[(i + 2) & 3], s3 = sn[(i + 2) & 3];
            const float c4 = cs[(i + 3) & 3], s4 = sn[(i + 3) & 3];
            // w = RZ * RY|0> ; z = RX * w ; u = RY * z
            const Cpx w0 = { c1 * c2, -c1 * s2 };
            const Cpx w1 = { s1 * c2,  s1 * s2 };
            const Cpx z0 = { c3 * w0.re + s3 * w1.im, c3 * w0.im - s3 * w1.re };
            const Cpx z1 = { s3 * w0.im + c3 * w1.re, -s3 * w0.re + c3 * w1.im };
            u[i][0] = { c4 * z0.re - s4 * z1.re, c4 * z0.im - s4 * z1.im };
            u[i][1] = { s4 * z0.re + c4 * z1.re, s4 * z0.im + c4 * z1.im };
        }

        Cpx c3sel;
        c3sel.re = h ? u[3][1].re : u[3][0].re;
        c3sel.im = h ? u[3][1].im : u[3][0].im;

        Cpx t4[4];
#pragma unroll
        for (int m = 0; m < 4; ++m) t4[m] = cmul(u[0][m >> 1], u[1][m & 1]);

        Cpx sfeat[8];
#pragma unroll
        for (int kc = 0; kc < 8; ++kc)
            sfeat[kc] = cmul(cmul(t4[kc >> 1], u[2][kc & 1]), c3sel);

        // ---- WMMA: Yt = Wt x St ----
        v8f d0 = {};
        v8f d1 = {};
#pragma unroll
        for (int kc = 0; kc < 8; ++kc) {
            v2f bf;
            bf.x = sfeat[kc].re;
            bf.y = sfeat[kc].im;
            d0 = __builtin_amdgcn_wmma_f32_16x16x4_f32(
                false, aw[0][kc], false, bf, (short)0, d0, false, false);
            d1 = __builtin_amdgcn_wmma_f32_16x16x4_f32(
                false, aw[1][kc], false, bf, (short)0, d1, false, false);
        }

        // ---- measurement: |y|^2 with signs ----
        // lane's probability slots: k = 8*t + 4*h + v/2  (v = even VGPR idx)
        float p[8];
#pragma unroll
        for (int v = 0; v < 8; v += 2) {
            p[v >> 1]       = d0[v] * d0[v] + d0[v + 1] * d0[v + 1];
            p[4 + (v >> 1)] = d1[v] * d1[v] + d1[v + 1] * d1[v + 1];
        }
        // wire0 sign = bit3 = tile; wire1 = bit2 = h; wire2 = bit1; wire3 = bit0
        float q0 = (p[0] + p[1] + p[2] + p[3]) - (p[4] + p[5] + p[6] + p[7]);
        float sAll = p[0] + p[1] + p[2] + p[3] + p[4] + p[5] + p[6] + p[7];
        float q1 = h ? -sAll : sAll;
        float q2 = (p[0] + p[1] - p[2] - p[3]) + (p[4] + p[5] - p[6] - p[7]);
        float q3 = (p[0] - p[1] + p[2] - p[3]) + (p[4] - p[5] + p[6] - p[7]);

        q0 += __shfl_xor(q0, 16, 32);
        q1 += __shfl_xor(q1, 16, 32);
        q2 += __shfl_xor(q2, 16, 32);
        q3 += __shfl_xor(q3, 16, 32);

        if (h == 0) {
            float4 o; o.x = q0; o.y = q1; o.z = q2; o.w = q3;
            ((float4*)out)[batch] = o;
        }
    }
}

// ---------------------------------------------------------------------------
extern "C" void kernel_launch(void* const* d_in, const int* in_sizes, int n_in,
                              void* d_out, int out_size, void* d_ws, size_t ws_size,
                              hipStream_t stream) {
    const float* x      = (const float*)d_in[0];  // (B, 4) float32
    const float* params = (const float*)d_in[1];  // (3, 4, 3) float32
    float* W   = (float*)d_ws;                    // 32x32 real matrix (4 KB)
    float* out = (float*)d_out;                   // (B, 4) float32

    const int B = in_sizes[0] / N_WIRES;
    const int numTiles = B / 16;

    hipLaunchKernelGGL(build_W_kernel, dim3(1), dim3(32), 0, stream, params, W);

    int blocks = (numTiles + 7) / 8;
    if (blocks > 2048) blocks = 2048;
    hipLaunchKernelGGL(qsim_kernel, dim3(blocks), dim3(256), 0, stream,
                       x, W, out, numTiles);
}